// RelPos_89996744721177
// MI455X (gfx1250) — compile-verified
//
#include <hip/hip_runtime.h>
#include <hip/hip_bf16.h>
#include <stdint.h>

// ---------------------------------------------------------------------------
// RelPos gather: out[i,j,c] = Wp_weight[c, ResInd[i,j]] + Wp_bias[c]
// SEQ=2048, CZ=64, VBINS=4095. Output = 1 GiB fp32 -> pure store-BW bound.
// ---------------------------------------------------------------------------

#define SEQ   2048
#define CZ    64
#define VBINS 4095

typedef __attribute__((ext_vector_type(4))) float v4f;
typedef __attribute__((ext_vector_type(4))) int   v4i;

#define AS1 __attribute__((address_space(1)))
#define AS3 __attribute__((address_space(3)))

#if defined(__gfx1250__) && __has_builtin(__builtin_amdgcn_global_load_async_to_lds_b128)
#define HAVE_ASYNC_LDS 1
#else
#define HAVE_ASYNC_LDS 0
#endif

__device__ __forceinline__ void wait_asynccnt0() {
#if __has_builtin(__builtin_amdgcn_s_wait_asynccnt)
  __builtin_amdgcn_s_wait_asynccnt(0);
#else
  asm volatile("s_wait_asynccnt 0x0" ::: "memory");
#endif
}

// ---------------------------------------------------------------------------
// Kernel 1: tableB[idx*64 + c] = Wp_weight[c*VBINS + idx] + bias[c]
// (folds transpose AND bias: main kernel then does zero FP math)
// ---------------------------------------------------------------------------
__global__ void relpos_prep(const float* __restrict__ W,
                            const float* __restrict__ bias,
                            float* __restrict__ tableB) {
  const int idx = blockIdx.x * blockDim.x + threadIdx.x;
  if (idx >= VBINS) return;
#pragma unroll 8
  for (int c = 0; c < CZ; ++c) {
    // read coalesced across lanes (fixed c, consecutive idx)
    tableB[idx * CZ + c] = W[(size_t)c * VBINS + idx] + bias[c];
  }
}

// ---------------------------------------------------------------------------
// Kernel 2: main gather.
// 256 threads/block; 16 lanes per (i,j) pair (one float4 each -> 256B row);
// 8 iterations -> 128 pairs/block.
// ---------------------------------------------------------------------------
#define THREADS 256
#define ITERS   8
#define PAIRS_PER_ITER  (THREADS / 16)           // 16
#define PAIRS_PER_BLOCK (PAIRS_PER_ITER * ITERS) // 128

__global__ __launch_bounds__(THREADS) void relpos_gather(
    const int*   __restrict__ ResInd,
    const float* __restrict__ tableB,
    float*       __restrict__ out) {
  const int t   = threadIdx.x;
  const int c4  = t & 15;    // which float4 of the 64-float row
  const int prl = t >> 4;    // pair-slot within an iteration (0..15)
  const long long pBase = (long long)blockIdx.x * PAIRS_PER_BLOCK + prl;

  // gather the 8 indices for this thread's pairs first
  int idxs[ITERS];
#pragma unroll
  for (int it = 0; it < ITERS; ++it) {
    idxs[it] = ResInd[pBase + (long long)it * PAIRS_PER_ITER];
  }

#if HAVE_ASYNC_LDS
  // Stage gathered rows via async global->LDS DMA (ASYNCcnt-tracked),
  // then drain LDS -> nontemporal 128-bit global stores.
  __shared__ __align__(16) float stage[ITERS][THREADS * 4];  // 32 KB

#pragma unroll
  for (int it = 0; it < ITERS; ++it) {
    const float* src = tableB + (long long)idxs[it] * CZ + (c4 << 2);
    __builtin_amdgcn_global_load_async_to_lds_b128(
        (AS1 v4i*)(v4i*)(void*)src,
        (AS3 v4i*)(v4i*)(void*)&stage[it][t << 2],
        0, 0);
  }
  wait_asynccnt0();

#pragma unroll
  for (int it = 0; it < ITERS; ++it) {
    const long long p = pBase + (long long)it * PAIRS_PER_ITER;
    v4f v = *(const v4f*)&stage[it][t << 2];
    __builtin_nontemporal_store(v, (v4f*)(out + p * CZ + (c4 << 2)));
  }
#else
  // Fallback: direct 128-bit gather loads (table is L2-resident, rows are
  // 256B so 16 lanes x 16B is one fully-coalesced row), NT 128-bit stores
  // so the 1 GiB write stream does not thrash L2.
#pragma unroll
  for (int it = 0; it < ITERS; ++it) {
    const long long p = pBase + (long long)it * PAIRS_PER_ITER;
    v4f v = *(const v4f*)(tableB + (long long)idxs[it] * CZ + (c4 << 2));
    __builtin_nontemporal_store(v, (v4f*)(out + p * CZ + (c4 << 2)));
  }
#endif
}

// ---------------------------------------------------------------------------
// Host-side launcher
// ---------------------------------------------------------------------------
extern "C" void kernel_launch(void* const* d_in, const int* in_sizes, int n_in,
                              void* d_out, int out_size, void* d_ws, size_t ws_size,
                              hipStream_t stream) {
  const int*   ResInd = (const int*)d_in[0];    // (2048, 2048) int32
  const float* W      = (const float*)d_in[1];  // (64, 4095) f32
  const float* bias   = (const float*)d_in[2];  // (64,) f32
  float*       out    = (float*)d_out;          // (2048, 2048, 64) f32
  float*       tableB = (float*)d_ws;           // 4095*64 f32 = ~1 MB scratch

  // 1) fold transpose + bias into tableB
  relpos_prep<<<(VBINS + 255) / 256, 256, 0, stream>>>(W, bias, tableB);

  // 2) gather + stream out
  const int nPairs = SEQ * SEQ;                       // 4,194,304
  const int nBlocks = nPairs / PAIRS_PER_BLOCK;       // 32,768
  relpos_gather<<<nBlocks, THREADS, 0, stream>>>(ResInd, tableB, out);
}